// Predictor_8924942041234
// MI455X (gfx1250) — compile-verified
//
#include <hip/hip_runtime.h>
#include <hip/hip_bf16.h>

#define N_ENT   20000
#define N_REL   24
#define N_EDGE  200000
#define BATCH   64
#define N_RULES 32

typedef __attribute__((ext_vector_type(2))) float v2f;
typedef __attribute__((ext_vector_type(8))) float v8f;

// ---------------- workspace layout (bytes) ----------------
// degcur : int[N_ENT]     @ 0        (histogram, then scatter cursors)
// off    : int[N_ENT+1]   @ 80000
// csr    : int[N_EDGE]    @ 160016
// keep   : uchar[N_EDGE]  @ 960016
#define WS_DEGCUR 0
#define WS_OFF    80000
#define WS_CSR    160016
#define WS_KEEP   960016

// 1) keep[e]=1, degcur[i]=0
__global__ void k_init(unsigned char* keep, int* degcur) {
    int i = blockIdx.x * blockDim.x + threadIdx.x;
    if (i < N_EDGE) keep[i] = 1;
    if (i < N_ENT)  degcur[i] = 0;
}

// 2) drop removed edges
__global__ void k_remove(const int* edges_to_remove, unsigned char* keep) {
    int t = threadIdx.x;
    if (t < BATCH) {
        int e = edges_to_remove[t];
        if (e >= 0 && e < N_EDGE) keep[e] = 0;
    }
}

// 3) histogram edges by head
__global__ void k_hist(const int* heads, int* degcur) {
    int e = blockIdx.x * blockDim.x + threadIdx.x;
    if (e < N_EDGE) atomicAdd(&degcur[heads[e]], 1);
}

// 4) single-WG exclusive scan of degrees -> off[], cursors re-seeded in degcur[]
#define SCAN_T 1024
#define CHUNK  20   // 1024*20 = 20480 >= N_ENT
__global__ void k_scan(int* degcur, int* off) {
    __shared__ int partA[SCAN_T];
    __shared__ int partB[SCAN_T];
    int t = threadIdx.x;
    int begin = t * CHUNK;
    int sum = 0;
    for (int i = 0; i < CHUNK; ++i) {
        int idx = begin + i;
        if (idx < N_ENT) sum += degcur[idx];
    }
    partA[t] = sum;
    __syncthreads();
    int* src = partA; int* dst = partB;
    for (int o = 1; o < SCAN_T; o <<= 1) {
        int v = src[t];
        if (t >= o) v += src[t - o];
        dst[t] = v;
        __syncthreads();
        int* tmp = src; src = dst; dst = tmp;
    }
    int running = (t == 0) ? 0 : src[t - 1];
    for (int i = 0; i < CHUNK; ++i) {
        int idx = begin + i;
        if (idx < N_ENT) {
            int d = degcur[idx];
            off[idx]    = running;
            degcur[idx] = running;   // cursor for scatter
            running += d;
        }
    }
    if (t == 0) off[N_ENT] = src[SCAN_T - 1];
}

// 5) scatter edge ids into CSR
__global__ void k_scatter(const int* heads, int* cursor, int* csr) {
    int e = blockIdx.x * blockDim.x + threadIdx.x;
    if (e < N_EDGE) {
        int pos = atomicAdd(&cursor[heads[e]], 1);
        csr[pos] = e;
    }
}

// 6) output init: score[b,:] = bias (via WMMA broadcast), mask = 1.0
//    blocks [0,625): 8 waves each -> 5000 waves, one 16x16 tile per wave.
//    blocks [625,1875): fill mask region with 1.0f (float4 stores).
__global__ void k_initout(const float* __restrict__ bias, float* __restrict__ out,
                          int out_size) {
    if (blockIdx.x < 625) {
        int lane = threadIdx.x & 31;
        int wave = blockIdx.x * (blockDim.x >> 5) + (threadIdx.x >> 5); // 0..4999
        int rowbase = (wave & 3) << 4;        // 0,16,32,48
        int colbase = (wave >> 2) << 4;       // 0..19984, step 16 (1250 tiles)
        int n = lane & 15;
        float bval = bias[colbase + n];

        // A: 16x4, only column K=0 is ones  -> lanes 0-15 hold K=0,K=1
        v2f a; a.x = (lane < 16) ? 1.0f : 0.0f; a.y = 0.0f;
        // B: 4x16, only row K=0 is bias     -> lanes 0-15 hold K=0,K=1
        v2f b; b.x = (lane < 16) ? bval : 0.0f; b.y = 0.0f;
        v8f c = {};
        v8f d = __builtin_amdgcn_wmma_f32_16x16x4_f32(
            /*neg_a=*/false, a, /*neg_b=*/false, b,
            /*c_mod=*/(short)0, c, /*reuse_a=*/false, /*reuse_b=*/false);

        int mbase = (lane < 16) ? 0 : 8;
        #pragma unroll
        for (int i = 0; i < 8; ++i) {
            int m = rowbase + mbase + i;
            out[(size_t)m * N_ENT + colbase + n] = d[i];
        }
    } else {
        // mask region: BATCH*N_ENT floats of 1.0f after the score region
        if (out_size < 2 * BATCH * N_ENT) return;
        size_t j = (size_t)(blockIdx.x - 625) * blockDim.x + threadIdx.x; // 0..319999
        float4* mask4 = (float4*)(out + (size_t)BATCH * N_ENT);
        mask4[j] = make_float4(1.0f, 1.0f, 1.0f, 1.0f);
    }
}

// 7) sparse 2-hop rule grounding: one block per (rule, batch-row)
__global__ void k_traverse(const int* __restrict__ all_h,
                           const int* __restrict__ rels,
                           const int* __restrict__ tails,
                           const int* __restrict__ rbodies,
                           const float* __restrict__ rweights,
                           const unsigned char* __restrict__ keep,
                           const int* __restrict__ off,
                           const int* __restrict__ csr,
                           float* __restrict__ score) {
    int k = blockIdx.x >> 6;   // rule
    int b = blockIdx.x & 63;   // batch row
    int r0 = rbodies[2 * k];
    int r1 = rbodies[2 * k + 1];
    float w = rweights[k];
    int h = all_h[b];
    int s = off[h], e = off[h + 1];
    float* srow = score + (size_t)b * N_ENT;
    for (int i = s + threadIdx.x; i < e; i += blockDim.x) {
        int e1 = csr[i];
        if (rels[e1] != r0 || !keep[e1]) continue;
        int m = tails[e1];
        int s2 = off[m], e2n = off[m + 1];
        for (int j = s2; j < e2n; ++j) {
            int e2 = csr[j];
            if (rels[e2] != r1 || !keep[e2]) continue;
            atomicAdd(&srow[tails[e2]], w);
        }
    }
}

extern "C" void kernel_launch(void* const* d_in, const int* in_sizes, int n_in,
                              void* d_out, int out_size, void* d_ws, size_t ws_size,
                              hipStream_t stream) {
    const int*   all_h    = (const int*)d_in[0];
    // d_in[1] = all_r (unused by reference math)
    const int*   removes  = (const int*)d_in[2];
    const int*   heads    = (const int*)d_in[3];
    const int*   rels     = (const int*)d_in[4];
    const int*   tails    = (const int*)d_in[5];
    const int*   rbodies  = (const int*)d_in[6];
    const float* rweights = (const float*)d_in[7];
    const float* bias     = (const float*)d_in[8];
    float* out = (float*)d_out;

    char* ws = (char*)d_ws;
    int*           degcur = (int*)(ws + WS_DEGCUR);
    int*           off    = (int*)(ws + WS_OFF);
    int*           csr    = (int*)(ws + WS_CSR);
    unsigned char* keep   = (unsigned char*)(ws + WS_KEEP);

    const int EB = (N_EDGE + 255) / 256;   // 782

    k_init    <<<EB, 256, 0, stream>>>(keep, degcur);
    k_remove  <<<1, BATCH, 0, stream>>>(removes, keep);
    k_hist    <<<EB, 256, 0, stream>>>(heads, degcur);
    k_scan    <<<1, SCAN_T, 0, stream>>>(degcur, off);
    k_scatter <<<EB, 256, 0, stream>>>(heads, degcur, csr);
    k_initout <<<1875, 256, 0, stream>>>(bias, out, out_size);
    k_traverse<<<N_RULES * BATCH, 64, 0, stream>>>(all_h, rels, tails, rbodies,
                                                   rweights, keep, off, csr, out);
}